// RotaryEmbedding_64699387347495
// MI455X (gfx1250) — compile-verified
//
#include <hip/hip_runtime.h>

typedef float v2f __attribute__((ext_vector_type(2)));
typedef float v8f __attribute__((ext_vector_type(8)));

#ifndef __has_builtin
#define __has_builtin(x) 0
#endif

// Positional-encoding kernel (store-bandwidth bound: 4 MiB in -> 512 MiB out,
// ~23 us floor at 23.3 TB/s).
//
//   angle_norm[n, i*64+m] = x[n,i] * 10000^(-m/64)        (angle in "turns")
//   out[n, 2k]   = cos(2*pi*angle) + sin(2*pi*angle)
//   out[n, 2k+1] = cos(2*pi*angle) - sin(2*pi*angle)
//
// v_sin_f32 / v_cos_f32 natively take normalized (turns) input, so the 2*pi
// scale in the reference cancels exactly.
//
// One wave handles 16 consecutive positions. The 16x256 angle matrix is
// computed as A(16x4) * W(4x256) via 16 V_WMMA_F32_16X16X4_F32 ops
// (W block-diagonal: W[K, i*64+m] = invf[m] if K==i else 0).
// Tile loop is unrolled only 4x to keep <=4 WMMA accumulators (32 VGPRs) live
// instead of 16 (128 VGPRs), preserving occupancy for store-latency hiding.
__global__ __launch_bounds__(256) void RotaryEmbedding_gfx1250_kernel(
    const float* __restrict__ x, float* __restrict__ out, int n_waves) {
  const int lane  = threadIdx.x & 31;
  const int wave  = (blockIdx.x << 3) + (threadIdx.x >> 5);
  if (wave >= n_waves) return;          // never taken for exact grid (EXEC stays full)
  const int n_base = wave << 4;         // first of 16 positions for this wave

  const int col  = lane & 15;           // tile column / A-matrix row M
  const int half = lane >> 4;           // 0: K=0,1 ; 1: K=2,3

  // ---- A matrix (16x4 f32, 2 VGPRs): lane<16 -> coords 0,1 ; lane>=16 -> 2,3
  const v2f* xv = (const v2f*)x;        // x rows are 4 floats = 2 x v2f
  v2f a = xv[((n_base + col) << 1) + half];

  // ---- normalized inverse frequencies: invf[m] = 2^(-m * log2(10000)/64)
  const float kNegL2T = -0.20762050593f;  // -log2(10000)/64
  float invf4[4];
#pragma unroll
  for (int q = 0; q < 4; ++q)
    invf4[q] = __builtin_exp2f((float)((q << 4) + col) * kNegL2T);

  v2f* outv = (v2f*)out;
  const int kbase = half << 1;

#pragma unroll 4
  for (int t = 0; t < 16; ++t) {        // 16 column tiles of 16 -> 256 angle cols
    const int   i_t = t >> 2;           // which coordinate feeds this tile
    const float fv  = invf4[t & 3];     // invf for m = (t&3)*16 + col

    v8f ang;
#if __has_builtin(__builtin_amdgcn_wmma_f32_16x16x4_f32)
    // B matrix (4x16 f32, 2 VGPRs): lane<16 holds K=0,1; lane>=16 holds K=2,3;
    // column N = lane%16. Block-diagonal selector x frequency.
    v2f b;
    b[0] = (kbase + 0 == i_t) ? fv : 0.0f;
    b[1] = (kbase + 1 == i_t) ? fv : 0.0f;
    v8f cz = {};
    ang = __builtin_amdgcn_wmma_f32_16x16x4_f32(
        /*neg_a=*/false, a, /*neg_b=*/false, b,
        /*c_mod=*/(short)0, cz, /*reuse_a=*/false, /*reuse_b=*/false);
#else
    // Scalar fallback: same output mapping, direct multiply.
#pragma unroll
    for (int r = 0; r < 8; ++r) {
      const int row = n_base + r + (half << 3);
      ang[r] = x[(row << 2) + i_t] * fv;
    }
#endif

#pragma unroll
    for (int r = 0; r < 8; ++r) {
      const int row = n_base + r + (half << 3);       // D-matrix row for this VGPR
      const float s = __builtin_amdgcn_sinf(ang[r]);  // sin(2*pi*ang)
      const float c = __builtin_amdgcn_cosf(ang[r]);  // cos(2*pi*ang)
      v2f o; o[0] = c + s; o[1] = c - s;
      // float2 index: row*256 + tile*16 + col -> lanes 0-15 and 16-31 each form
      // one contiguous 128B segment; non-temporal (512 MB >> 192 MB L2).
      __builtin_nontemporal_store(o, &outv[(row << 8) + (t << 4) + col]);
    }
  }
}

extern "C" void kernel_launch(void* const* d_in, const int* in_sizes, int n_in,
                              void* d_out, int out_size, void* d_ws, size_t ws_size,
                              hipStream_t stream) {
  (void)n_in; (void)out_size; (void)d_ws; (void)ws_size;
  const float* x = (const float*)d_in[0];
  float* out = (float*)d_out;

  const int positions = in_sizes[0] / 4;        // (B*N) rows of 4 coords
  const int n_waves   = positions / 16;         // 16 positions per wave
  const int blocks    = (n_waves + 7) / 8;      // 8 waves (256 threads) per block

  RotaryEmbedding_gfx1250_kernel<<<blocks, 256, 0, stream>>>(x, out, n_waves);
}